// EventExtractor_69363721831095
// MI455X (gfx1250) — compile-verified
//
#include <hip/hip_runtime.h>

typedef __attribute__((ext_vector_type(2))) float v2f;
typedef __attribute__((ext_vector_type(8))) float v8f;

// ---------------- workspace layout (float offsets) ----------------
#define OFF_X      0L
#define OFF_XWF    131072L
#define OFF_XWB    655360L
#define OFF_SEQ    1179648L
#define OFF_POSL   1441792L
#define OFF_NERL   1492992L
#define OFF_TRIGL  1513472L
#define OFF_PREDP  1554432L
#define OFF_PREDN  1555456L
#define OFF_NERC   1556480L
#define OFF_TRIGC  1557504L
#define OFF_FUSED  1558528L
#define OFF_EA     2009088L
#define OFF_EAW    2025472L
#define OFF_DEG    2156544L
#define OFF_NPART  2157568L
#define OFF_SELFP  2288640L
#define OFF_AGG    2419712L
#define OFF_H1     2550784L
#define OFF_H2     2681856L
#define OFF_QB     2812928L
#define OFF_KB     2944000L
#define OFF_ATTN   3075072L
#define OFF_BILHW  3206144L
#define OFF_BIL    3337216L
#define OFF_P1A    3468288L
#define OFF_P2A    3599360L
#define OFF_P1R    3730432L
#define OFF_P2R    3861504L
#define OFF_DISTW  3992576L

__device__ __forceinline__ v8f wmma4(v2f a, v2f b, v8f c) {
  return __builtin_amdgcn_wmma_f32_16x16x4_f32(false, a, false, b, (short)0, c, false, false);
}

// ---------------- generic fp32 WMMA GEMM: C = alpha*(A @ op(W)) + bias, opt ReLU ---
// transB=0: op(W)=W^T with W[N,K] (element W[n*ldw+k]);  transB=1: W[K,N] (element W[k*ldw+n])
// A/B tiles staged through LDS in K-chunks of 64 -> inner loop is ds_load + v_wmma only.
__global__ void __launch_bounds__(128) gemm_k(
    const float* __restrict__ A, int lda, long strA,
    const float* __restrict__ W, int ldw, long strW, int transB,
    const float* __restrict__ bias, float* __restrict__ C,
    int ldc, long strC, int M, int Nn, int K, int dorelu, float alpha)
{
  __shared__ float sAt[16][64];
  __shared__ float sBt[64][64];
  const float* Ab = A + (long)blockIdx.z * strA;
  const float* Wb = W + (long)blockIdx.z * strW;
  float* Cb = C + (long)blockIdx.z * strC;
  int wave = threadIdx.x >> 5, lane = threadIdx.x & 31;
  int half = lane >> 4, r = lane & 15;
  int m0 = blockIdx.x * 16;
  int n0b = blockIdx.y * 64;
  int nl = wave * 16 + r;          // local n within the 64-wide block tile
  int nB = n0b + nl;
  v8f acc = {0.f,0.f,0.f,0.f,0.f,0.f,0.f,0.f};

  for (int kb0 = 0; kb0 < K; kb0 += 64) {
    int chunk = K - kb0; if (chunk > 64) chunk = 64;
    // stage A tile [16 x chunk]
    for (int e = threadIdx.x; e < 16 * 64; e += 128) {
      int row = e >> 6, col = e & 63;
      int m = m0 + row, k = kb0 + col;
      sAt[row][col] = (m < M && col < chunk) ? Ab[(long)m * lda + k] : 0.f;
    }
    // stage B tile [64 x chunk]
    for (int e = threadIdx.x; e < 64 * 64; e += 128) {
      int nrow = e >> 6, col = e & 63;
      int n = n0b + nrow, k = kb0 + col;
      float v = 0.f;
      if (n < Nn && col < chunk)
        v = transB ? Wb[(long)k * ldw + n] : Wb[(long)n * ldw + k];
      sBt[nrow][col] = v;
    }
    __syncthreads();
#pragma unroll 4
    for (int kk = 0; kk < chunk; kk += 4) {
      int k0 = kk + 2 * half;
      v2f a;  a.x  = sAt[r][k0];  a.y  = sAt[r][k0 + 1];
      v2f bf; bf.x = sBt[nl][k0]; bf.y = sBt[nl][k0 + 1];
      acc = wmma4(a, bf, acc);
    }
    __syncthreads();
  }

  if (nB < Nn) {
    float bv = bias ? bias[nB] : 0.f;
#pragma unroll
    for (int rr = 0; rr < 8; ++rr) {
      int m = m0 + rr + 8 * half;
      if (m < M) {
        float v = acc[rr] * alpha + bv;
        if (dorelu) v = fmaxf(v, 0.f);
        Cb[(long)m * ldc + nB] = v;
      }
    }
  }
}

// ---------------- embedding gather ----------------
__global__ void gather_x_k(const int* __restrict__ ids, const float* __restrict__ Ew,
                           float* __restrict__ x) {
  int tok = blockIdx.x;
  x[(long)tok * 128 + threadIdx.x] = Ew[(long)ids[tok] * 128 + threadIdx.x];
}

// ---------------- LSTM recurrence (XW precomputed) ----------------
__global__ void lstm_k(const float* __restrict__ XWf, const float* __restrict__ XWb,
                       const float* __restrict__ WhhF, const float* __restrict__ WhhB,
                       const int* __restrict__ lengths, float* __restrict__ seq)
{
  int dir = blockIdx.x, b = blockIdx.y, j = threadIdx.x;
  const float* XW  = dir ? XWb  : XWf;
  const float* Whh = dir ? WhhB : WhhF;
  __shared__ float hs[128];
  float c = 0.f; hs[j] = 0.f; __syncthreads();
  int len = lengths[b];
  for (int s = 0; s < 128; ++s) {
    int t = dir ? (127 - s) : s;
    long base = ((long)b * 128 + t) * 512;
    float zi = XW[base + j], zf = XW[base + 128 + j];
    float zg = XW[base + 256 + j], zo = XW[base + 384 + j];
    const float* wi = Whh + (long)j * 128;
    const float* wf = Whh + (long)(128 + j) * 128;
    const float* wg = Whh + (long)(256 + j) * 128;
    const float* wo = Whh + (long)(384 + j) * 128;
    for (int k = 0; k < 128; ++k) {
      float hk = hs[k];
      zi += wi[k] * hk; zf += wf[k] * hk; zg += wg[k] * hk; zo += wo[k] * hk;
    }
    float ig = 1.f / (1.f + expf(-zi));
    float fg = 1.f / (1.f + expf(-zf));
    float og = 1.f / (1.f + expf(-zo));
    c = fg * c + ig * tanhf(zg);
    float hn = og * tanhf(c);
    __syncthreads(); hs[j] = hn; __syncthreads();
    seq[((long)b * 128 + t) * 256 + dir * 128 + j] = (t < len) ? hn : 0.f;
  }
}

// ---------------- argmax + max-softmax over small class dim ----------------
__global__ void class_reduce_k(const float* __restrict__ logits, int C,
                               int* __restrict__ pred, float* __restrict__ conf)
{
  int tok = blockIdx.x, t = threadIdx.x; // 64 threads
  float v = (t < C) ? logits[(long)tok * C + t] : -1e30f;
  __shared__ float sv[64]; __shared__ int si[64]; __shared__ float ss[64];
  sv[t] = v; si[t] = t; __syncthreads();
  for (int off = 32; off > 0; off >>= 1) {
    if (t < off) {
      float ov = sv[t + off]; int oi = si[t + off];
      if (ov > sv[t] || (ov == sv[t] && oi < si[t])) { sv[t] = ov; si[t] = oi; }
    }
    __syncthreads();
  }
  float vmax = sv[0]; int imax = si[0]; __syncthreads();
  ss[t] = (t < C) ? expf(v - vmax) : 0.f; __syncthreads();
  for (int off = 32; off > 0; off >>= 1) { if (t < off) ss[t] += ss[t + off]; __syncthreads(); }
  if (t == 0) { if (pred) pred[tok] = imax; if (conf) conf[tok] = 1.0f / ss[0]; }
}

// ---------------- fused feature concat (440) ----------------
__global__ void build_fused_k(const float* __restrict__ seq, const int* __restrict__ ids,
                              const int* __restrict__ affix, const int* __restrict__ cases,
                              const int* __restrict__ predp, const int* __restrict__ predn,
                              const float* __restrict__ Eg, const float* __restrict__ Ea,
                              const float* __restrict__ Ep, const float* __restrict__ En,
                              const float* __restrict__ Ec, const float* __restrict__ morph,
                              float* __restrict__ fused)
{
  int tok = blockIdx.x;
  int wid = ids[tok], af = affix[tok], cs = cases[tok], pp = predp[tok], pn = predn[tok];
  for (int d = threadIdx.x; d < 440; d += blockDim.x) {
    float v;
    if      (d < 256) v = seq[(long)tok * 256 + d];
    else if (d < 320) v = Eg[(long)wid * 64 + (d - 256)];
    else if (d < 352) v = Ea[(long)af * 32 + (d - 320)];
    else if (d < 384) v = Ep[pp * 32 + (d - 352)];
    else if (d < 416) v = En[pn * 32 + (d - 384)];
    else if (d < 432) v = Ec[cs * 16 + (d - 416)];
    else              v = morph[(long)tok * 8 + (d - 432)];
    fused[(long)tok * 440 + d] = v;
  }
}

// ---------------- adj-weighted edge-emb reduction + degree ----------------
__global__ void edge_agg_k(const float* __restrict__ adj, const int* __restrict__ et,
                           const float* __restrict__ Eedge, float* __restrict__ ea,
                           float* __restrict__ deg)
{
  int tok = blockIdx.x, j = threadIdx.x;
  float a = adj[(long)tok * 128 + j];
  int e = et[(long)tok * 128 + j];
  float pe[16];
#pragma unroll
  for (int q = 0; q < 16; ++q) pe[q] = a * Eedge[e * 16 + q];
  __shared__ float red[128];
  red[j] = a; __syncthreads();
  for (int off = 64; off > 0; off >>= 1) { if (j < off) red[j] += red[j + off]; __syncthreads(); }
  if (j == 0) deg[tok] = red[0];
  for (int q = 0; q < 16; ++q) {
    __syncthreads();
    red[j] = pe[q]; __syncthreads();
    for (int off = 64; off > 0; off >>= 1) { if (j < off) red[j] += red[j + off]; __syncthreads(); }
    if (j == 0) ea[tok * 16 + q] = red[0];
  }
}

// ---------------- RGCN combine: relu(selfp + agg + eaW + deg*bm) ----------------
__global__ void combine_k(const float* __restrict__ selfp, const float* __restrict__ agg,
                          const float* __restrict__ eaW, const float* __restrict__ deg,
                          const float* __restrict__ bm, float* __restrict__ h)
{
  int t = blockIdx.x, n = threadIdx.x;
  long idx = (long)t * 128 + n;
  h[idx] = fmaxf(selfp[idx] + agg[idx] + eaW[idx] + deg[t] * bm[n], 0.f);
}

// ---------------- row softmax (len 128) ----------------
__global__ void softmax_rows_k(float* __restrict__ buf) {
  int row = blockIdx.x, t = threadIdx.x;
  __shared__ float sh[128];
  float v = buf[(long)row * 128 + t];
  sh[t] = v; __syncthreads();
  for (int off = 64; off > 0; off >>= 1) { if (t < off) sh[t] = fmaxf(sh[t], sh[t + off]); __syncthreads(); }
  float m = sh[0]; __syncthreads();
  float e = expf(v - m);
  sh[t] = e; __syncthreads();
  for (int off = 64; off > 0; off >>= 1) { if (t < off) sh[t] += sh[t + off]; __syncthreads(); }
  buf[(long)row * 128 + t] = e / sh[0];
}

// ---------------- dist-embedding rows through W1[:,512:528] ----------------
__global__ void prep_dist_k(const float* __restrict__ Ed, const float* __restrict__ argW,
                            const float* __restrict__ roleW, float* __restrict__ distW) {
  int n = threadIdx.x;
  for (int head = 0; head < 2; ++head) {
    const float* W1 = head ? roleW : argW;
    for (int k = 0; k < 9; ++k) {
      float s = 0.f;
      for (int e = 0; e < 16; ++e) s += Ed[k * 16 + e] * W1[(long)n * 535 + 512 + e];
      distW[(head * 9 + k) * 128 + n] = s;
    }
  }
}

// ---------------- fused pairwise MLP (WMMA over K=256 pairwise block) ----------------
__global__ void __launch_bounds__(256) pair_k(
    const float* __restrict__ h, const float* __restrict__ adj,
    const int* __restrict__ cases,
    const float* __restrict__ attn, const float* __restrict__ bil,
    const float* __restrict__ bil_b,
    const float* __restrict__ trigc, const float* __restrict__ nerc,
    const float* __restrict__ P1a, const float* __restrict__ P2a,
    const float* __restrict__ P1r, const float* __restrict__ P2r,
    const float* __restrict__ argW, const float* __restrict__ roleW,
    const float* __restrict__ distW,
    const float* __restrict__ arg2W, const float* __restrict__ arg2b,
    const float* __restrict__ role2W, const float* __restrict__ role2b,
    float* __restrict__ out)
{
  int i = blockIdx.x, jt = blockIdx.y, b = blockIdx.z;
  int j0 = jt * 16;
  int tid = threadIdx.x;
  __shared__ float shi[128];
  __shared__ float shj[16][128];
  __shared__ float sy[2][16][128];
  if (tid < 128) shi[tid] = h[((long)b * 128 + i) * 128 + tid];
  for (int idx = tid; idx < 16 * 128; idx += 256) {
    int jj = idx >> 7, k = idx & 127;
    shj[jj][k] = h[((long)b * 128 + j0 + jj) * 128 + k];
  }
  __syncthreads();

  int wave = tid >> 5, lane = tid & 31;
  int half = lane >> 4, r = lane & 15;
  int head = wave >> 2;                       // 0 = arg path, 1 = role path
  const float* W1 = head ? roleW : argW;
  const float* P1 = head ? P1r : P1a;
  const float* P2 = head ? P2r : P2a;
  int n0 = (wave & 3) * 32;
  int na = n0 + r, nb = n0 + 16 + r;
  v8f acc0 = {0,0,0,0,0,0,0,0}, acc1 = {0,0,0,0,0,0,0,0};
#pragma unroll 4
  for (int kb = 0; kb < 256; kb += 4) {
    int k0 = kb + 2 * half;
    float a0, a1;
    if (k0 < 128) {
      a0 = fabsf(shi[k0]     - shj[r][k0]);
      a1 = fabsf(shi[k0 + 1] - shj[r][k0 + 1]);
    } else {
      a0 = shi[k0 - 128] * shj[r][k0 - 128];
      a1 = shi[k0 - 127] * shj[r][k0 - 127];
    }
    v2f a; a.x = a0; a.y = a1;
    const float* wr0 = W1 + (long)na * 535 + 256 + k0;
    const float* wr1 = W1 + (long)nb * 535 + 256 + k0;
    v2f bfa; bfa.x = wr0[0]; bfa.y = wr0[1];
    v2f bfb; bfb.x = wr1[0]; bfb.y = wr1[1];
    acc0 = wmma4(a, bfa, acc0);
    acc1 = wmma4(a, bfb, acc1);
  }
  float wa[7], wb[7];
#pragma unroll
  for (int s = 0; s < 7; ++s) {
    wa[s] = W1[(long)na * 535 + 528 + s];
    wb[s] = W1[(long)nb * 535 + 528 + s];
  }
  float tg = trigc[b * 128 + i];
  float bbias = bil_b[0];
  long rowij = ((long)b * 128 + i) * 128;
  long p1row = ((long)b * 128 + i) * 128;
#pragma unroll
  for (int rr = 0; rr < 8; ++rr) {
    int jj = rr + 8 * half;
    int j = j0 + jj;
    float adjv = adj[rowij + j];
    float selfv = (i == j) ? 1.f : 0.f;
    float attnv = attn[rowij + j];
    float bilv = bil[rowij + j] + bbias;
    float nc = nerc[b * 128 + j];
    float cc = (cases[b * 128 + j] != 0) ? 1.f : 0.f;
    int d = j - i; d = d > 4 ? 4 : (d < -4 ? -4 : d);
    int sd = d + 4;
    long p2row = ((long)b * 128 + j) * 128;
    float ea = adjv*wa[0] + selfv*wa[1] + attnv*wa[2] + bilv*wa[3] + tg*wa[4] + nc*wa[5] + cc*wa[6];
    float va = acc0[rr] + ea + distW[(head * 9 + sd) * 128 + na] + P1[p1row + na] + P2[p2row + na];
    sy[head][jj][na] = fmaxf(va, 0.f);
    float eb = adjv*wb[0] + selfv*wb[1] + attnv*wb[2] + bilv*wb[3] + tg*wb[4] + nc*wb[5] + cc*wb[6];
    float vb = acc1[rr] + eb + distW[(head * 9 + sd) * 128 + nb] + P1[p1row + nb] + P2[p2row + nb];
    sy[head][jj][nb] = fmaxf(vb, 0.f);
  }
  __syncthreads();
  // second layer: 16 pairs x (2 arg + 30 role) outputs
  for (int s = 0; s < 2; ++s) {
    int idx = tid * 2 + s;
    int jj = idx >> 5, ch = idx & 31;
    const float* w; const float* y; float bo;
    if (ch < 2) { w = arg2W + ch * 128;        y = sy[0][jj]; bo = arg2b[ch]; }
    else        { w = role2W + (ch - 2) * 128; y = sy[1][jj]; bo = role2b[ch - 2]; }
    float sum = bo;
    for (int k = 0; k < 128; ++k) sum += w[k] * y[k];
    out[(((long)b * 128 + i) * 128 + (j0 + jj)) * 32 + ch] = sum;
  }
}

extern "C" void kernel_launch(void* const* d_in, const int* in_sizes, int n_in,
                              void* d_out, int out_size, void* d_ws, size_t ws_size,
                              hipStream_t stream) {
  (void)in_sizes; (void)n_in; (void)out_size; (void)ws_size;
  // inputs (setup_inputs order; params flattened in sorted-key pytree order)
  const int*   input_ids = (const int*)  d_in[0];
  const int*   lengths   = (const int*)  d_in[1];
  const float* adj       = (const float*)d_in[2];
  const int*   edge_type = (const int*)  d_in[3];
  const int*   affix     = (const int*)  d_in[4];
  const int*   caseids   = (const int*)  d_in[5];
  const float* morph     = (const float*)d_in[6];
  const float* E_affix   = (const float*)d_in[7];
  const float* E_case    = (const float*)d_in[8];
  const float* E_dist    = (const float*)d_in[9];
  const float* E_edge    = (const float*)d_in[10];
  const float* E_graph   = (const float*)d_in[11];
  const float* E_ner     = (const float*)d_in[12];
  const float* E_pos     = (const float*)d_in[13];
  const float* E_word    = (const float*)d_in[14];
  const float* Wk        = (const float*)d_in[15];
  const float* Wq        = (const float*)d_in[16];
  const float* arg1W     = (const float*)d_in[17];
  const float* arg1b     = (const float*)d_in[18];
  const float* arg2W     = (const float*)d_in[19];
  const float* arg2b     = (const float*)d_in[20];
  const float* bilW      = (const float*)d_in[21];
  const float* bil_b     = (const float*)d_in[22];
  const float* bk        = (const float*)d_in[23];
  const float* bq        = (const float*)d_in[24];
  const float* WhhB      = (const float*)d_in[25];
  const float* WihB      = (const float*)d_in[26];
  const float* bB        = (const float*)d_in[27];
  const float* WhhF      = (const float*)d_in[28];
  const float* WihF      = (const float*)d_in[29];
  const float* bF        = (const float*)d_in[30];
  const float* nerW      = (const float*)d_in[31];
  const float* nerb      = (const float*)d_in[32];
  const float* posW      = (const float*)d_in[33];
  const float* posb      = (const float*)d_in[34];
  const float* Wme0      = (const float*)d_in[35];
  const float* Wmn0      = (const float*)d_in[36];
  const float* Ws0       = (const float*)d_in[37];
  const float* bm0       = (const float*)d_in[38];
  const float* bs0       = (const float*)d_in[39];
  const float* Wme1      = (const float*)d_in[40];
  const float* Wmn1      = (const float*)d_in[41];
  const float* Ws1       = (const float*)d_in[42];
  const float* bm1       = (const float*)d_in[43];
  const float* bs1       = (const float*)d_in[44];
  const float* role1W    = (const float*)d_in[45];
  const float* role1b    = (const float*)d_in[46];
  const float* role2W    = (const float*)d_in[47];
  const float* role2b    = (const float*)d_in[48];
  const float* trigW     = (const float*)d_in[49];
  const float* trigb     = (const float*)d_in[50];

  float* wsf   = (float*)d_ws;
  float* x     = wsf + OFF_X;     float* XWf   = wsf + OFF_XWF;   float* XWb = wsf + OFF_XWB;
  float* seq   = wsf + OFF_SEQ;   float* posl  = wsf + OFF_POSL;  float* nerl = wsf + OFF_NERL;
  float* trigl = wsf + OFF_TRIGL;
  int*   predp = (int*)(wsf + OFF_PREDP);
  int*   predn = (int*)(wsf + OFF_PREDN);
  float* nercf = wsf + OFF_NERC;  float* trigcf = wsf + OFF_TRIGC;
  float* fused = wsf + OFF_FUSED; float* ea    = wsf + OFF_EA;    float* eaW = wsf + OFF_EAW;
  float* deg   = wsf + OFF_DEG;   float* npart = wsf + OFF_NPART; float* selfp = wsf + OFF_SELFP;
  float* agg   = wsf + OFF_AGG;   float* h1    = wsf + OFF_H1;    float* h2 = wsf + OFF_H2;
  float* Qb    = wsf + OFF_QB;    float* Kb    = wsf + OFF_KB;    float* attnb = wsf + OFF_ATTN;
  float* bilHW = wsf + OFF_BILHW; float* bilv  = wsf + OFF_BIL;
  float* P1a   = wsf + OFF_P1A;   float* P2a   = wsf + OFF_P2A;
  float* P1r   = wsf + OFF_P1R;   float* P2r   = wsf + OFF_P2R;
  float* distW = wsf + OFF_DISTW;
  float* outf  = (float*)d_out;

  auto gemm = [&](const float* A, int lda, long sA, const float* W, int ldw, long sW,
                  int transB, const float* bias, float* C, int ldc, long sC,
                  int M, int Nn, int K, int dorelu, float alpha, int batch) {
    dim3 g((unsigned)((M + 15) / 16), (unsigned)((Nn + 63) / 64), (unsigned)batch);
    gemm_k<<<g, 128, 0, stream>>>(A, lda, sA, W, ldw, sW, transB, bias, C, ldc, sC,
                                  M, Nn, K, dorelu, alpha);
  };

  // 1) embeddings + BiLSTM
  gather_x_k<<<1024, 128, 0, stream>>>(input_ids, E_word, x);
  gemm(x, 128, 0, WihF, 128, 0, 0, bF, XWf, 512, 0, 1024, 512, 128, 0, 1.f, 1);
  gemm(x, 128, 0, WihB, 128, 0, 0, bB, XWb, 512, 0, 1024, 512, 128, 0, 1.f, 1);
  lstm_k<<<dim3(2, 8), 128, 0, stream>>>(XWf, XWb, WhhF, WhhB, lengths, seq);

  // 2) pos / ner heads
  gemm(seq, 256, 0, posW, 256, 0, 0, posb, posl, 50, 0, 1024, 50, 256, 0, 1.f, 1);
  class_reduce_k<<<1024, 64, 0, stream>>>(posl, 50, predp, nullptr);
  gemm(seq, 256, 0, nerW, 256, 0, 0, nerb, nerl, 20, 0, 1024, 20, 256, 0, 1.f, 1);
  class_reduce_k<<<1024, 64, 0, stream>>>(nerl, 20, predn, nercf);

  // 3) fused features + edge aggregation
  build_fused_k<<<1024, 128, 0, stream>>>(seq, input_ids, affix, caseids, predp, predn,
                                          E_graph, E_affix, E_pos, E_ner, E_case, morph, fused);
  edge_agg_k<<<1024, 128, 0, stream>>>(adj, edge_type, E_edge, ea, deg);

  // 4) RGCN layer 1
  gemm(ea, 16, 0, Wme0, 16, 0, 0, nullptr, eaW, 128, 0, 1024, 128, 16, 0, 1.f, 1);
  gemm(fused, 440, 0, Wmn0, 440, 0, 0, nullptr, npart, 128, 0, 1024, 128, 440, 0, 1.f, 1);
  gemm(fused, 440, 0, Ws0, 440, 0, 0, bs0, selfp, 128, 0, 1024, 128, 440, 0, 1.f, 1);
  gemm(adj, 128, 16384, npart, 128, 16384, 1, nullptr, agg, 128, 16384, 128, 128, 128, 0, 1.f, 8);
  combine_k<<<1024, 128, 0, stream>>>(selfp, agg, eaW, deg, bm0, h1);

  // 5) RGCN layer 2
  gemm(ea, 16, 0, Wme1, 16, 0, 0, nullptr, eaW, 128, 0, 1024, 128, 16, 0, 1.f, 1);
  gemm(h1, 128, 0, Wmn1, 128, 0, 0, nullptr, npart, 128, 0, 1024, 128, 128, 0, 1.f, 1);
  gemm(h1, 128, 0, Ws1, 128, 0, 0, bs1, selfp, 128, 0, 1024, 128, 128, 0, 1.f, 1);
  gemm(adj, 128, 16384, npart, 128, 16384, 1, nullptr, agg, 128, 16384, 128, 128, 128, 0, 1.f, 8);
  combine_k<<<1024, 128, 0, stream>>>(selfp, agg, eaW, deg, bm1, h2);

  // 6) trigger confidence
  gemm(h2, 128, 0, trigW, 128, 0, 0, trigb, trigl, 40, 0, 1024, 40, 128, 0, 1.f, 1);
  class_reduce_k<<<1024, 64, 0, stream>>>(trigl, 40, nullptr, trigcf);

  // 7) attention
  gemm(h2, 128, 0, Wq, 128, 0, 0, bq, Qb, 128, 0, 1024, 128, 128, 0, 1.f, 1);
  gemm(h2, 128, 0, Wk, 128, 0, 0, bk, Kb, 128, 0, 1024, 128, 128, 0, 1.f, 1);
  gemm(Qb, 128, 16384, Kb, 128, 16384, 0, nullptr, attnb, 128, 16384,
       128, 128, 128, 0, 0.08838834764831845f, 8);
  softmax_rows_k<<<1024, 128, 0, stream>>>(attnb);

  // 8) bilinear
  gemm(h2, 128, 0, bilW, 128, 0, 1, nullptr, bilHW, 128, 0, 1024, 128, 128, 0, 1.f, 1);
  gemm(bilHW, 128, 16384, h2, 128, 16384, 0, nullptr, bilv, 128, 16384, 128, 128, 128, 0, 1.f, 8);

  // 9) pairwise precompute (hi / hj halves of W1, biases folded into P1)
  gemm(h2, 128, 0, arg1W, 535, 0, 0, arg1b, P1a, 128, 0, 1024, 128, 128, 0, 1.f, 1);
  gemm(h2, 128, 0, arg1W + 128, 535, 0, 0, nullptr, P2a, 128, 0, 1024, 128, 128, 0, 1.f, 1);
  gemm(h2, 128, 0, role1W, 535, 0, 0, role1b, P1r, 128, 0, 1024, 128, 128, 0, 1.f, 1);
  gemm(h2, 128, 0, role1W + 128, 535, 0, 0, nullptr, P2r, 128, 0, 1024, 128, 128, 0, 1.f, 1);
  prep_dist_k<<<1, 128, 0, stream>>>(E_dist, arg1W, role1W, distW);

  // 10) fused pair MLP -> output [B,N,N,32]
  pair_k<<<dim3(128, 8, 8), 256, 0, stream>>>(h2, adj, caseids, attnb, bilv, bil_b,
      trigcf, nercf, P1a, P2a, P1r, P2r, arg1W, role1W, distW,
      arg2W, arg2b, role2W, role2b, outf);
}